// SparseDownBlock3d_v1_82429012345629
// MI455X (gfx1250) — compile-verified
//
#include <hip/hip_runtime.h>
#include <hip/hip_bf16.h>

typedef float v2f __attribute__((ext_vector_type(2)));
typedef float v8f __attribute__((ext_vector_type(8)));

#define C_CH 256
#define TILE_M 16
#define LDS_STRIDE 260   // 260 floats = 1040 B; 1040/4 % 64 = 4 banks/row -> conflict-free

__device__ __forceinline__ int voxel_seg(const int* __restrict__ coords, int n,
                                         int factor, int g) {
    int bb = coords[n * 4 + 0];
    int x  = coords[n * 4 + 1] / factor;
    int y  = coords[n * 4 + 2] / factor;
    int z  = coords[n * 4 + 3] / factor;
    return ((bb * g + x) * g + y) * g + z;
}

// ---- pass 1: per-voxel point counts --------------------------------------
__global__ void sdb3d_count_kernel(const int* __restrict__ coords,
                                   const int* __restrict__ dFactor,
                                   const int* __restrict__ dGrid,
                                   float* __restrict__ cnt, int nPoints) {
    int n = blockIdx.x * blockDim.x + threadIdx.x;
    if (n >= nPoints) return;
    int f = *dFactor;
    int g = (*dGrid) / f;
    int s = voxel_seg(coords, n, f, g);
    unsafeAtomicAdd(&cnt[s], 1.0f);
}

// ---- pass 2: WMMA fp32 GEMM + bias + SiLU + segment scatter-add ----------
__global__ __launch_bounds__(512)
void sdb3d_gemm_silu_scatter(const float* __restrict__ feats,
                             const float* __restrict__ W,
                             const float* __restrict__ bias,
                             const int* __restrict__ coords,
                             const int* __restrict__ dFactor,
                             const int* __restrict__ dGrid,
                             float* __restrict__ outSums, int nPoints) {
    __shared__ float As[TILE_M * LDS_STRIDE];
    __shared__ int   segs[TILE_M];

    const int t     = threadIdx.x;
    const int wave  = t >> 5;      // 0..15 -> N-tile (output-channel tile)
    const int lane  = t & 31;
    const int mbase = blockIdx.x * TILE_M;

    // Cooperative A-tile load: 512 threads x 8 contiguous floats = 16x256.
    {
        int idx = t * 8;               // 0..4095
        int row = idx >> 8;            // 0..15
        int col = idx & 255;           // multiple of 8 -> 32B aligned
        float4 a0 = make_float4(0.f, 0.f, 0.f, 0.f);
        float4 a1 = a0;
        if (mbase + row < nPoints) {
            const float* src = feats + (size_t)(mbase + row) * C_CH + col;
            a0 = *(const float4*)(src);
            a1 = *(const float4*)(src + 4);
        }
        float* dst = &As[row * LDS_STRIDE + col];
        *(float4*)(dst)     = a0;
        *(float4*)(dst + 4) = a1;
    }
    if (t < TILE_M) {
        int n = mbase + t;
        int s = -1;
        if (n < nPoints) {
            int f = *dFactor;
            int g = (*dGrid) / f;
            s = voxel_seg(coords, n, f, g);
        }
        segs[t] = s;
    }
    __syncthreads();

    // Fragment addressing per ISA 16x4 f32 layout:
    //   lane L holds M = L&15; K offset = 2*(L>>4) + {0,1} within each K=4 step.
    const int halfSel = lane >> 4;
    const int lrow    = lane & 15;
    const int o       = wave * 16 + lrow;                 // output channel (B: N=o)
    const float* wrow = W + (size_t)o * C_CH + 2 * halfSel; // B[K=c][N=o] = W[o][c]
    const float* arow = &As[lrow * LDS_STRIDE + 2 * halfSel];

    v8f acc = {};
#pragma unroll 8
    for (int k = 0; k < C_CH / 4; ++k) {
        v2f afrag = *(const v2f*)(arow + 4 * k);   // ds_load_2addr_b64, conflict-free
        v2f bfrag = *(const v2f*)(wrow + 4 * k);   // global_load_b64, L2-resident
        acc = __builtin_amdgcn_wmma_f32_16x16x4_f32(
            /*neg_a=*/false, afrag, /*neg_b=*/false, bfrag,
            /*c_mod=*/(short)0, acc, /*reuse_a=*/false, /*reuse_b=*/false);
    }

    // C/D layout: VGPR r holds row M = r + 8*(L>>4), column N = L&15 (fixed o/lane).
    const float bo = bias[o];
#pragma unroll
    for (int r = 0; r < 8; ++r) {
        int m = r + 8 * halfSel;
        int s = segs[m];
        if (s >= 0) {
            float v = acc[r] + bo;
            // SiLU via raw v_rcp_f32 (avoid IEEE divide expansion)
            v = v * __builtin_amdgcn_rcpf(1.0f + __expf(-v));
            unsafeAtomicAdd(&outSums[(size_t)s * C_CH + o], v);
        }
    }
}

// ---- pass 3: divide sums by max(count,1) ---------------------------------
__global__ void sdb3d_finalize_kernel(float* __restrict__ out,
                                      const float* __restrict__ cnt,
                                      int outSize) {
    int total4 = outSize >> 2;
    for (int i = blockIdx.x * blockDim.x + threadIdx.x; i < total4;
         i += gridDim.x * blockDim.x) {
        float4 v = ((float4*)out)[i];
        int seg  = (i << 2) >> 8;   // 256 ch per segment; float4 never crosses rows
        float inv = __builtin_amdgcn_rcpf(fmaxf(cnt[seg], 1.0f));
        v.x *= inv; v.y *= inv; v.z *= inv; v.w *= inv;
        ((float4*)out)[i] = v;
    }
}

extern "C" void kernel_launch(void* const* d_in, const int* in_sizes, int n_in,
                              void* d_out, int out_size, void* d_ws, size_t ws_size,
                              hipStream_t stream) {
    const float* feats  = (const float*)d_in[0];
    const float* W      = (const float*)d_in[1];
    const float* bias   = (const float*)d_in[2];
    const int*   coords = (const int*)d_in[3];
    const int*   dF     = (const int*)d_in[4];   // factor
    const int*   dG     = (const int*)d_in[5];   // grid_size
    // d_in[6] = batch_size (num_seg already implied by out_size)

    const int nPoints = in_sizes[0] / C_CH;
    const int numSeg  = out_size / C_CH;

    float* outSums = (float*)d_out;
    float* cnt     = (float*)d_ws;

    // zero accumulators (graph-capturable memset nodes)
    hipMemsetAsync(outSums, 0, (size_t)out_size * sizeof(float), stream);
    hipMemsetAsync(cnt, 0, (size_t)numSeg * sizeof(float), stream);

    // pass 1: counts
    {
        int blocks = (nPoints + 255) / 256;
        sdb3d_count_kernel<<<blocks, 256, 0, stream>>>(coords, dF, dG, cnt, nPoints);
    }
    // pass 2: GEMM + SiLU + scatter
    {
        int blocks = (nPoints + TILE_M - 1) / TILE_M;
        sdb3d_gemm_silu_scatter<<<blocks, 512, 0, stream>>>(
            feats, W, bias, coords, dF, dG, outSums, nPoints);
    }
    // pass 3: mean
    {
        int total4 = out_size / 4;
        int blocks = (total4 + 255) / 256;
        if (blocks > 65535) blocks = 65535;
        sdb3d_finalize_kernel<<<blocks, 256, 0, stream>>>(outSums, cnt, out_size);
    }
}